// GATNE_T_54863912239204
// MI455X (gfx1250) — compile-verified
//
#include <hip/hip_runtime.h>
#include <math.h>

// GATNE-T fused kernel for MI455X (gfx1250, wave32).
// Shapes: V=500000, T=4, D=32, E=128, A=32, B=8192, S=10.

typedef __attribute__((ext_vector_type(2))) float v2f;
typedef __attribute__((ext_vector_type(8))) float v8f;

#define TT 4
#define DD 32
#define EE 128
#define AA 32
#define SS 10
#define BBLK 16          // batch rows per block
#define NCOLS (TT*EE)    // 512: proj computed against all 4 type weights

__global__ __launch_bounds__(256) void gatne_fused_kernel(
    const int*   __restrict__ targets,          // [B]
    const int*   __restrict__ types,            // [B]
    const int*   __restrict__ neighbors,        // [B,T,S]
    const float* __restrict__ base_emb,         // [V,E]
    const float* __restrict__ nte,              // [V,T,D]
    const float* __restrict__ tw,               // [T,D,E]
    const float* __restrict__ tw1,              // [T,D,A]
    const float* __restrict__ tw2,              // [T,A]
    float*       __restrict__ out)              // [B,E]
{
    __shared__ float s_tw1[TT * DD * AA];       // 16 KB
    __shared__ float s_tw2[TT * AA];            // 512 B
    __shared__ float s_natt[BBLK][DD + 1];      // A-tile, padded (stride 33)
    __shared__ int   s_type[BBLK];
    __shared__ float s_C[BBLK][NCOLS];          // 32 KB proj result (all types)

    const int tid  = threadIdx.x;
    const int wave = tid >> 5;
    const int lane = tid & 31;
    const int b0   = blockIdx.x * BBLK;

    // ---- stage weights for attention into LDS ----
    for (int i = tid; i < TT * DD * AA; i += 256) s_tw1[i] = tw1[i];
    for (int i = tid; i < TT * AA;      i += 256) s_tw2[i] = tw2[i];
    __syncthreads();

    // ---- phase 1: gather + mean + attention; each wave owns 2 rows ----
    #pragma unroll
    for (int rr = 0; rr < 2; ++rr) {
        const int r  = wave * 2 + rr;
        const int b  = b0 + r;
        const int ty = types[b];

        // neighbor gather + mean: lane = d, coalesced 128B per (t,s) row
        float agg[TT];
        #pragma unroll
        for (int t = 0; t < TT; ++t) {
            float acc = 0.f;
            const int* nb = neighbors + ((size_t)b * TT + t) * SS;
            #pragma unroll
            for (int s = 0; s < SS; ++s) {
                const int n = nb[s];
                acc += nte[(size_t)n * (TT * DD) + t * DD + lane];
            }
            agg[t] = acc * (1.0f / SS);
        }

        // u = tanh(agg @ W1[ty]); score[t] = reduce(u * W2[ty])
        const float* w1 = s_tw1 + ty * (DD * AA);
        const float* w2 = s_tw2 + ty * AA;
        float score[TT];
        #pragma unroll
        for (int t = 0; t < TT; ++t) {
            float u = 0.f;
            #pragma unroll
            for (int d = 0; d < DD; ++d) {
                u = fmaf(__shfl(agg[t], d, 32), w1[d * AA + lane], u);
            }
            u = tanhf(u);
            float p = u * w2[lane];
            #pragma unroll
            for (int off = 16; off > 0; off >>= 1) p += __shfl_xor(p, off, 32);
            score[t] = p;   // all lanes hold the full score
        }

        // softmax over T=4 (replicated across lanes)
        const float m = fmaxf(fmaxf(score[0], score[1]), fmaxf(score[2], score[3]));
        float ex[TT]; float esum = 0.f;
        #pragma unroll
        for (int t = 0; t < TT; ++t) { ex[t] = __expf(score[t] - m); esum += ex[t]; }
        const float inv = 1.0f / esum;

        // node_att[d] = sum_t att[t] * agg[t][d]  (lane = d)
        float natt = 0.f;
        #pragma unroll
        for (int t = 0; t < TT; ++t) natt = fmaf(ex[t] * inv, agg[t], natt);

        s_natt[r][lane] = natt;
        if (lane == 0) s_type[r] = ty;
    }
    __syncthreads();

    // ---- phase 2: proj GEMM via v_wmma_f32_16x16x4_f32 ----
    // C[16 x 512] = node_att[16 x 32] @ W_all[32 x 512], W_all[d][t*128+e] = tw[t][d][e]
    // Each wave computes 4 N-tiles of 16 cols; K=32 as 8 chained k=4 WMMAs.
    #pragma unroll
    for (int i = 0; i < 4; ++i) {
        const int ntile   = wave * 4 + i;
        const int colBase = ntile * 16;
        const int col     = colBase + (lane & 15);
        const int ty      = col >> 7;          // which type's weight slab
        const int e       = col & (EE - 1);
        const float* Wt   = tw + (size_t)ty * (DD * EE) + e;  // index with [k*EE]
        const int row     = lane & 15;
        const int khalf   = (lane >> 4) << 1;  // lanes 0-15: K 0,1; lanes 16-31: K 2,3

        v8f acc = {};
        #pragma unroll
        for (int kk = 0; kk < 8; ++kk) {
            const int k = kk * 4 + khalf;
            v2f a, bfrag;
            a.x     = s_natt[row][k];
            a.y     = s_natt[row][k + 1];
            bfrag.x = Wt[(size_t)k * EE];
            bfrag.y = Wt[(size_t)(k + 1) * EE];
            acc = __builtin_amdgcn_wmma_f32_16x16x4_f32(
                false, a, false, bfrag, (short)0, acc, false, false);
        }
        // C layout: VGPR j -> M = j + 8*(lane>=16), N = lane&15
        const int mOff = (lane >> 4) * 8;
        const int n    = lane & 15;
        #pragma unroll
        for (int j = 0; j < 8; ++j) {
            s_C[mOff + j][colBase + n] = acc[j];
        }
    }
    __syncthreads();

    // ---- phase 3: epilogue — slice by type, add base row, L2 normalize ----
    #pragma unroll
    for (int rr = 0; rr < 2; ++rr) {
        const int r  = wave * 2 + rr;
        const int b  = b0 + r;
        const int ty = s_type[r];
        const float* cRow  = &s_C[r][ty * EE];
        const float* basep = base_emb + (size_t)targets[b] * EE;

        float v[4]; float sq = 0.f;
        #pragma unroll
        for (int i = 0; i < 4; ++i) {
            const int e = lane + 32 * i;
            v[i] = cRow[e] + basep[e];
            sq = fmaf(v[i], v[i], sq);
        }
        #pragma unroll
        for (int off = 16; off > 0; off >>= 1) sq += __shfl_xor(sq, off, 32);
        const float rinv = rsqrtf(fmaxf(sq, 1e-12f));
        #pragma unroll
        for (int i = 0; i < 4; ++i) {
            out[(size_t)b * EE + lane + 32 * i] = v[i] * rinv;
        }
    }
}

extern "C" void kernel_launch(void* const* d_in, const int* in_sizes, int n_in,
                              void* d_out, int out_size, void* d_ws, size_t ws_size,
                              hipStream_t stream) {
    (void)in_sizes; (void)n_in; (void)d_ws; (void)ws_size; (void)out_size;
    const int*   targets   = (const int*)  d_in[0];
    const int*   types     = (const int*)  d_in[1];
    const int*   neighbors = (const int*)  d_in[2];
    const float* base_emb  = (const float*)d_in[3];
    const float* nte       = (const float*)d_in[4];
    const float* tw        = (const float*)d_in[5];
    const float* tw1       = (const float*)d_in[6];
    const float* tw2       = (const float*)d_in[7];
    float* out = (float*)d_out;

    const int B = 8192;
    dim3 grid(B / BBLK);   // 512 blocks, 16 batch rows each
    dim3 block(256);       // 8 wave32
    gatne_fused_kernel<<<grid, block, 0, stream>>>(
        targets, types, neighbors, base_emb, nte, tw, tw1, tw2, out);
}